// SummaryBothAttention_38706245271873
// MI455X (gfx1250) — compile-verified
//
#include <hip/hip_runtime.h>
#include <math.h>

#define B_ 8
#define T_ 32
#define S_ 256
#define D_ 1024

typedef __attribute__((ext_vector_type(2))) float v2f;
typedef __attribute__((ext_vector_type(8))) float v8f;

// Native gfx1250 transcendental tanh if the toolchain exposes it.
#if __has_builtin(__builtin_amdgcn_tanhf)
  #define FAST_TANH(x) __builtin_amdgcn_tanhf(x)
#elif __has_builtin(__builtin_amdgcn_tanh_f32)
  #define FAST_TANH(x) __builtin_amdgcn_tanh_f32(x)
#else
  #define FAST_TANH(x) tanhf(x)
#endif

// ---------------------------------------------------------------------------
// Kernel 1: running prefix-mean over T, then mask.  s{1,2}[b,t,d]
// One block per b, one thread per d (coalesced), serial T=32 scan in regs.
// ---------------------------------------------------------------------------
__global__ void __launch_bounds__(1024) accum_kernel(
    const float* __restrict__ src1, const float* __restrict__ src2,
    const float* __restrict__ m1, const float* __restrict__ m2,
    float* __restrict__ s1, float* __restrict__ s2) {
  const int b = blockIdx.x;
  const int d = threadIdx.x;
  float r1 = 0.f, r2 = 0.f;
  for (int t = 0; t < T_; ++t) {
    size_t idx = ((size_t)b * T_ + t) * D_ + d;
    r1 += src1[idx];
    r2 += src2[idx];
    const float inv = 1.0f / (float)(t + 1);
    s1[idx] = r1 * inv * m1[b * T_ + t];
    s2[idx] = r2 * inv * m2[b * T_ + t];
  }
}

// ---------------------------------------------------------------------------
// Kernel 2: wsum[d] = sum_j W_v[j, d]   (column sums, coalesced)
// ---------------------------------------------------------------------------
__global__ void __launch_bounds__(256) wsum_kernel(
    const float* __restrict__ Wv, float* __restrict__ wsum) {
  const int d = blockIdx.x * blockDim.x + threadIdx.x;
  float s = 0.f;
  for (int j = 0; j < D_; ++j) s += Wv[(size_t)j * D_ + d];
  wsum[d] = s;
}

// ---------------------------------------------------------------------------
// Kernel 3: pa[b,s,c] = mem[b,s,:] . W_cate[c, :D]
//           pb[b,s,c] = mem[b,s,:] . W_cate[c, D:]
// Wave-per-(b,s), lanes stride D (coalesced), butterfly reduce.
// ---------------------------------------------------------------------------
__global__ void __launch_bounds__(256) pab_kernel(
    const float* __restrict__ mem, const float* __restrict__ Wc,
    float* __restrict__ pa, float* __restrict__ pb) {
  const int lane = threadIdx.x & 31;
  const int gw = blockIdx.x * (blockDim.x >> 5) + (threadIdx.x >> 5);
  const int b = gw / S_, s = gw % S_;
  const float* mrow = mem + ((size_t)b * S_ + s) * D_;
  float a0 = 0.f, a1 = 0.f, p0 = 0.f, p1 = 0.f;
  for (int d = lane; d < D_; d += 32) {
    const float m = mrow[d];
    a0 += m * Wc[d];                 // W_cate[0, d]
    a1 += m * Wc[2 * D_ + d];        // W_cate[1, d]
    p0 += m * Wc[D_ + d];            // W_cate[0, D + d]
    p1 += m * Wc[2 * D_ + D_ + d];   // W_cate[1, D + d]
  }
  for (int off = 16; off; off >>= 1) {
    a0 += __shfl_xor(a0, off);
    a1 += __shfl_xor(a1, off);
    p0 += __shfl_xor(p0, off);
    p1 += __shfl_xor(p1, off);
  }
  if (lane == 0) {
    pa[gw * 2 + 0] = a0;
    pa[gw * 2 + 1] = a1;
    pb[gw * 2 + 0] = p0;
    pb[gw * 2 + 1] = p1;
  }
}

// ---------------------------------------------------------------------------
// Kernel 4: q12 = s1 @ W_q1^T + s2 @ W_q2^T + (b_q1 + b_q2)    [256 x 1024]
// Native fp32 WMMA: V_WMMA_F32_16X16X4_F32, one wave per 16x16 output tile.
// A(16x4) lane layout: lanes 0-15 hold K={0,1}, lanes 16-31 hold K={2,3}.
// B(4x16)=W^T mirrored layout -> same per-lane indexing applied to W row.
// C/D: VGPR j -> rows j (lanes 0-15) and j+8 (lanes 16-31), col = n0+lid.
// ---------------------------------------------------------------------------
__global__ void __launch_bounds__(32) qproj_wmma_kernel(
    const float* __restrict__ s1, const float* __restrict__ s2,
    const float* __restrict__ Wq1, const float* __restrict__ bq1,
    const float* __restrict__ Wq2, const float* __restrict__ bq2,
    float* __restrict__ q12) {
  const int lane = threadIdx.x;
  const int lid  = lane & 15;
  const int half = lane >> 4;
  const int nt = blockIdx.x & 63;   // 1024/16 = 64 N-tiles
  const int mt = blockIdx.x >> 6;   // 256/16  = 16 M-tiles
  const int m0 = mt * 16, n0 = nt * 16;
  const int koff = 2 * half;

  const float* arow1 = s1  + (size_t)(m0 + lid) * D_ + koff;
  const float* arow2 = s2  + (size_t)(m0 + lid) * D_ + koff;
  const float* brow1 = Wq1 + (size_t)(n0 + lid) * D_ + koff;
  const float* brow2 = Wq2 + (size_t)(n0 + lid) * D_ + koff;

  v8f acc = {};
  for (int k0 = 0; k0 < D_; k0 += 4) {
    v2f a1 = *(const v2f*)(arow1 + k0);
    v2f b1 = *(const v2f*)(brow1 + k0);
    acc = __builtin_amdgcn_wmma_f32_16x16x4_f32(
        false, a1, false, b1, (short)0, acc, false, false);
    v2f a2 = *(const v2f*)(arow2 + k0);
    v2f b2 = *(const v2f*)(brow2 + k0);
    acc = __builtin_amdgcn_wmma_f32_16x16x4_f32(
        false, a2, false, b2, (short)0, acc, false, false);
  }

  const float bias = bq1[n0 + lid] + bq2[n0 + lid];
  const int col = n0 + lid;
#pragma unroll
  for (int j = 0; j < 8; ++j) {
    const int row = m0 + j + 8 * half;
    q12[(size_t)row * D_ + col] = acc[j] + bias;
  }
}

// ---------------------------------------------------------------------------
// Kernel 5 (dominant): for each (b,t,s)
//   sg = sum_d tanh(q12[b,t,d] + mem[b,s,d]) * wsum[d]
//   out[b,t,s,c] = pa[b,s,c] + sg * pb[b,s,c] + b_cate[c]
// Block per (b,t). The d-reduction is lane-assignment-agnostic, so each lane
// owns d = lane*4 + 128*i + {0..3}: q12/wsum register-cached (loaded once,
// reused for all 256 s), memory_bank streamed with global_load_b128. No LDS.
// ---------------------------------------------------------------------------
__global__ void __launch_bounds__(256) gate_kernel(
    const float* __restrict__ mem, const float* __restrict__ q12,
    const float* __restrict__ wsum, const float* __restrict__ pa,
    const float* __restrict__ pb, const float* __restrict__ bc,
    float* __restrict__ out) {
  const int bt = blockIdx.x;
  const int b = bt / T_;
  const int lane = threadIdx.x & 31;
  const int wv = threadIdx.x >> 5;
  const int base = lane * 4;  // this lane owns d = base + 128*i + {0..3}

  float4 q4[8], w4[8];
  const float* qrow = q12 + (size_t)bt * D_;
#pragma unroll
  for (int i = 0; i < 8; ++i) {
    q4[i] = *(const float4*)(qrow + base + 128 * i);
    w4[i] = *(const float4*)(wsum + base + 128 * i);
  }
  const float c0 = bc[0], c1 = bc[1];

  for (int s = wv; s < S_; s += 8) {
    const float* mrow = mem + ((size_t)b * S_ + s) * D_;
    if (s + 8 < S_)  // prefetch next row this wave will consume (L2 resident)
      __builtin_prefetch(mrow + 8 * D_ + lane * 32, 0, 3);
    float acc = 0.f;
#pragma unroll
    for (int i = 0; i < 8; ++i) {
      const float4 m = *(const float4*)(mrow + base + 128 * i);
      acc += FAST_TANH(q4[i].x + m.x) * w4[i].x;
      acc += FAST_TANH(q4[i].y + m.y) * w4[i].y;
      acc += FAST_TANH(q4[i].z + m.z) * w4[i].z;
      acc += FAST_TANH(q4[i].w + m.w) * w4[i].w;
    }
    for (int off = 16; off; off >>= 1) acc += __shfl_xor(acc, off);
    if (lane == 0) {
      const int bs = b * S_ + s;
      float2 o;
      o.x = pa[bs * 2 + 0] + acc * pb[bs * 2 + 0] + c0;
      o.y = pa[bs * 2 + 1] + acc * pb[bs * 2 + 1] + c1;
      *reinterpret_cast<float2*>(out + ((size_t)bt * S_ + s) * 2) = o;
    }
  }
}

// ---------------------------------------------------------------------------
// Workspace layout (floats):
//   s1   [B*T*D] = 262144 @ 0
//   s2   [B*T*D] = 262144 @ 262144
//   q12  [B*T*D] = 262144 @ 524288
//   pa   [B*S*2] =   4096 @ 786432
//   pb   [B*S*2] =   4096 @ 790528
//   wsum [D]     =   1024 @ 794624     (total ~3.04 MB)
// ---------------------------------------------------------------------------
extern "C" void kernel_launch(void* const* d_in, const int* in_sizes, int n_in,
                              void* d_out, int out_size, void* d_ws, size_t ws_size,
                              hipStream_t stream) {
  const float* source1  = (const float*)d_in[0];
  const float* source2  = (const float*)d_in[1];
  const float* mem      = (const float*)d_in[2];
  const float* masks1   = (const float*)d_in[3];
  const float* masks2   = (const float*)d_in[4];
  const float* W_q1     = (const float*)d_in[5];
  const float* b_q1     = (const float*)d_in[6];
  const float* W_q2     = (const float*)d_in[7];
  const float* b_q2     = (const float*)d_in[8];
  const float* W_cate   = (const float*)d_in[9];
  const float* b_cate   = (const float*)d_in[10];
  const float* W_v      = (const float*)d_in[11];
  float* out = (float*)d_out;

  float* w    = (float*)d_ws;
  float* s1   = w;
  float* s2   = w + 262144;
  float* q12  = w + 524288;
  float* pa   = w + 786432;
  float* pb   = w + 790528;
  float* wsum = w + 794624;

  accum_kernel<<<B_, 1024, 0, stream>>>(source1, source2, masks1, masks2, s1, s2);
  wsum_kernel<<<D_ / 256, 256, 0, stream>>>(W_v, wsum);
  pab_kernel<<<(B_ * S_) / 8, 256, 0, stream>>>(mem, W_cate, pa, pb);
  qproj_wmma_kernel<<<(B_ * T_ / 16) * (D_ / 16), 32, 0, stream>>>(
      s1, s2, W_q1, b_q1, W_q2, b_q2, q12);
  gate_kernel<<<B_ * T_, 256, 0, stream>>>(mem, q12, wsum, pa, pb, b_cate, out);
}